// LossYOLO_v3_76828374991002
// MI455X (gfx1250) — compile-verified
//
#include <hip/hip_runtime.h>
#include <math.h>

// ---- Problem constants (from reference) ----
#define B_      32
#define NGT_    20
#define HWA_    10647          // (2704+676+169)*3
#define NCLS_   80
#define BPB_    42             // blocks per batch: ceil(10647/256)
#define FP16_EPS_ 0.0009765625f
#define BCE_EPS_  1e-6f

typedef __attribute__((ext_vector_type(2))) float v2f;
typedef __attribute__((ext_vector_type(8))) float v8f;

__device__ __forceinline__ float sigmoidf_(float x) {
    return 1.0f / (1.0f + expf(-x));
}
__device__ __forceinline__ float clipp_(float p) {
    return fminf(fmaxf(p, BCE_EPS_), 1.0f - BCE_EPS_);
}

// ---------------- Kernel 0: zero the conf-state array ----------------
__global__ void k_zero(float* __restrict__ conf, int n) {
    int i = blockIdx.x * blockDim.x + threadIdx.x;
    if (i < n) conf[i] = 0.0f;
}

// ---------------- Kernel 1: sequential match replay (1 thread / batch) ----------------
// Faithfully replays the (buggy) NGT*NGT scan including write order:
// for each (i,k): anchor from GT k, box from GT i; per level j: write -1 to 3 rows,
// then (if ceng==j) overwrite the positive row.
__global__ void k_match(const float* __restrict__ gboxes,  // [B,NGT,4] ltrb
                        const int*   __restrict__ glabels, // [B,NGT]
                        float* __restrict__ conf,          // [B,HWA] state
                        float* __restrict__ posdata)       // [B,HWA,6] tx,ty,tw,th,w,label-1
{
    int b = threadIdx.x;
    if (b >= B_) return;

    const float ancw[9] = {0.024f, 0.038f, 0.079f, 0.072f, 0.149f, 0.142f, 0.279f, 0.375f, 0.897f};
    const float anch[9] = {0.031f, 0.072f, 0.055f, 0.147f, 0.108f, 0.286f, 0.216f, 0.476f, 0.784f};
    const int   GR[3]   = {52, 26, 13};
    const int   OF[3]   = {0, 2704, 3380};

    float cx[NGT_], cy[NGT_], gw[NGT_], gh[NGT_], wt[NGT_];
    int   ids[NGT_], lab[NGT_];

    const float* gb = gboxes + (size_t)b * NGT_ * 4;
    for (int k = 0; k < NGT_; ++k) {
        float l = gb[k*4+0], t = gb[k*4+1], r = gb[k*4+2], bt = gb[k*4+3];
        cx[k] = (l + r) * 0.5f;
        cy[k] = (t + bt) * 0.5f;
        gw[k] = r - l;
        gh[k] = bt - t;
        wt[k] = 2.0f - gw[k] * gh[k];
        lab[k] = glabels[b * NGT_ + k];
    }
    // argmax IoU(wh, anchor) over 9 anchors; first-max on ties (strict >)
    for (int k = 0; k < NGT_; ++k) {
        float best = -1.0f; int id = 0;
        for (int a = 0; a < 9; ++a) {
            float iw = fminf(gw[k], ancw[a]);
            float ih = fminf(gh[k], anch[a]);
            float inter = iw * ih;
            float uni = gw[k]*gh[k] + ancw[a]*anch[a] - inter;
            float r = inter / uni;
            if (r > best) { best = r; id = a; }
        }
        ids[k] = id;
    }

    float* cbase = conf    + (size_t)b * HWA_;
    float* pbase = posdata + (size_t)b * HWA_ * 6;

    for (int i = 0; i < NGT_; ++i) {
        for (int k = 0; k < NGT_; ++k) {
            int a = ids[k];
            int ceng = a / 3;
            for (int j = 0; j < 3; ++j) {
                int   g  = GR[j];
                float gf = (float)g;
                int col = (int)(cx[i] * gf);
                int row = (int)(cy[i] * gf);
                int idx = (OF[j] + row * g + col) * 3;
                cbase[idx + 0] = -1.0f;
                cbase[idx + 1] = -1.0f;
                cbase[idx + 2] = -1.0f;
                if (ceng == j) {
                    int pidx = idx + (a % 3);
                    cbase[pidx] = 1.0f;
                    float* pd = pbase + (size_t)pidx * 6;
                    pd[0] = (cx[i] - (float)col / gf) * gf;
                    pd[1] = (cy[i] - (float)row / gf) * gf;
                    pd[2] = logf(gw[i] / ancw[a]);
                    pd[3] = logf(gh[i] / anch[a]);
                    pd[4] = wt[i];
                    pd[5] = (float)(lab[i] - 1);
                }
            }
        }
    }
}

// ---------------- Kernel 2: main reduction, WMMA block reduce ----------------
// grid = (BPB_, B_), block = 256. Writes 7 partial sums per block.
__global__ void k_reduce(const float* __restrict__ pconf,   // [B,HWA]
                         const float* __restrict__ pcls,    // [B,HWA,80]
                         const float* __restrict__ ptxywh,  // [B,HWA,4]
                         const float* __restrict__ conf,    // [B,HWA]
                         const float* __restrict__ posdata, // [B,HWA,6]
                         float* __restrict__ partials)      // [B,BPB,7]
{
    const int b   = blockIdx.y;
    const int row = blockIdx.x * 256 + threadIdx.x;

    float acc[7] = {0.f, 0.f, 0.f, 0.f, 0.f, 0.f, 0.f};

    if (row < HWA_) {
        size_t rix = (size_t)b * HWA_ + row;
        float c = conf[rix];
        float s = sigmoidf_(pconf[rix]);
        float d = s - c;
        float lv = d * d;
        bool pos = (c > 0.0f);
        bool neg = (c == 0.0f);
        acc[0] = pos ? 1.0f : 0.0f;
        acc[1] = neg ? 1.0f : 0.0f;
        acc[2] = pos ? lv : 0.0f;
        acc[3] = neg ? lv : 0.0f;

        if (pos) {
            const float* pd = posdata + rix * 6;
            int L = (int)pd[5];
            // class BCE: onehot target
            const float* pc = pcls + rix * (size_t)NCLS_;
            float clssum = 0.0f;
            for (int cc = 0; cc < NCLS_; ++cc) {
                float p = clipp_(sigmoidf_(pc[cc]));
                clssum += (cc == L) ? -logf(p) : -logf(1.0f - p);
            }
            acc[4] = clssum;
            // txty BCE (weighted)
            const float* pw = ptxywh + rix * 4;
            float tsum = 0.0f;
            for (int dd = 0; dd < 2; ++dd) {
                float p = clipp_(sigmoidf_(pw[dd]));
                float g = pd[dd];
                tsum += -(g * logf(p) + (1.0f - g) * logf(1.0f - p));
            }
            acc[5] = pd[4] * tsum;
            // twth MSE (weighted)
            float e0 = pw[2] - pd[2];
            float e1 = pw[3] - pd[3];
            acc[6] = pd[4] * (e0 * e0 + e1 * e1);
        }
    }

    __shared__ float sd[7][256];
    for (int q = 0; q < 7; ++q) sd[q][threadIdx.x] = acc[q];
    __syncthreads();

    // Wave 0 reduces each accumulator's 256 values with 4 chained
    // v_wmma_f32_16x16x4_f32 ops against an all-ones B matrix.
    // A is 16x4 f32 = 64 values/wave: lane L holds A[L&15][2*(L>>4)] (VGPR0)
    // and A[L&15][2*(L>>4)+1] (VGPR1). D[m][n] = sum_k A[m][k] (+C).
    if (threadIdx.x < 32) {
        const int lane = threadIdx.x;
        const int m    = lane & 15;
        const int koff = (lane >> 4) * 2;
        v2f bones; bones.x = 1.0f; bones.y = 1.0f;

        for (int q = 0; q < 7; ++q) {
            v8f cacc = {};
            for (int part = 0; part < 4; ++part) {
                const float* src = &sd[q][part * 64];
                v2f a;
                a.x = src[m * 4 + koff];
                a.y = src[m * 4 + koff + 1];
                cacc = __builtin_amdgcn_wmma_f32_16x16x4_f32(
                    /*neg_a=*/false, a, /*neg_b=*/false, bones,
                    /*c_mod=*/(short)0, cacc, /*reuse_a=*/false, /*reuse_b=*/false);
            }
            // lanes 0-15 hold rows M=0..7 in cacc[0..7]; lanes 16-31 rows M=8..15
            float sred = cacc[0] + cacc[1] + cacc[2] + cacc[3] +
                         cacc[4] + cacc[5] + cacc[6] + cacc[7];
            sred += __shfl_xor(sred, 16, 32);
            if (lane == 0)
                partials[((size_t)b * BPB_ + blockIdx.x) * 7 + q] = sred;
        }
    }
}

// ---------------- Kernel 3: deterministic serial finalize ----------------
__global__ void k_finalize(const float* __restrict__ partials, float* __restrict__ out) {
    float cp = 0.f, cn = 0.f, ptot = 0.f, ntot = 0.f;
    float lc = 0.f, lt = 0.f, lw = 0.f;
    for (int b = 0; b < B_; ++b) {
        float pb = 0.f, clsb = 0.f, txb = 0.f, twb = 0.f;
        for (int blk = 0; blk < BPB_; ++blk) {
            const float* P = partials + ((size_t)b * BPB_ + blk) * 7;
            pb   += P[0];
            ntot += P[1];
            cp   += P[2];
            cn   += P[3];
            clsb += P[4];
            txb  += P[5];
            twb  += P[6];
        }
        ptot += pb;
        float nums = fmaxf(pb, FP16_EPS_);
        lc += clsb / nums;
        lt += txb  / nums;
        lw += twb  / nums;
    }
    float loss = cp / ptot * 30.0f + cn / ntot * 30.0f
               + (lc + lt + lw) / (float)B_;
    out[0] = loss;
}

// ---------------- Host launcher ----------------
extern "C" void kernel_launch(void* const* d_in, const int* in_sizes, int n_in,
                              void* d_out, int out_size, void* d_ws, size_t ws_size,
                              hipStream_t stream) {
    (void)in_sizes; (void)n_in; (void)out_size;

    const float* pconf   = (const float*)d_in[0];   // [B,HWA,1]
    const float* pcls    = (const float*)d_in[1];   // [B,HWA,80]
    const float* ptxywh  = (const float*)d_in[2];   // [B,HWA,4]
    const float* gboxes  = (const float*)d_in[3];   // [B,NGT,4]
    const int*   glabels = (const int*)d_in[4];     // [B,NGT]
    float* out = (float*)d_out;

    // workspace layout
    char* ws = (char*)d_ws;
    const size_t confN  = (size_t)B_ * HWA_;             // floats
    const size_t posN   = (size_t)B_ * HWA_ * 6;         // floats
    const size_t partN  = (size_t)B_ * BPB_ * 7;         // floats
    float* conf     = (float*)ws;
    float* posdata  = (float*)(ws + confN * sizeof(float));
    float* partials = (float*)(ws + (confN + posN) * sizeof(float));
    (void)ws_size; // requires ~9.6 MB; harness scratch is expected to cover this

    // 0) zero conf state
    {
        int n = (int)confN;
        int blocks = (n + 255) / 256;
        k_zero<<<blocks, 256, 0, stream>>>(conf, n);
    }
    // 1) sequential match replay (1 thread per batch)
    k_match<<<1, 32, 0, stream>>>(gboxes, glabels, conf, posdata);
    // 2) main reduction with WMMA block reduce
    {
        dim3 grid(BPB_, B_);
        k_reduce<<<grid, 256, 0, stream>>>(pconf, pcls, ptxywh, conf, posdata, partials);
    }
    // 3) deterministic finalize
    k_finalize<<<1, 1, 0, stream>>>(partials, out);
}